// JointLoss_74113955660001
// MI455X (gfx1250) — compile-verified
//
#include <hip/hip_runtime.h>
#include <hip/hip_bf16.h>
#include <math.h>

#define NB 8192
#define ND 768
#define APITCH 772   // 772 % 64 == 4 -> conflict-free LDS b64 reads
#define ROWS 32      // rows per block (two stacked 16x16 WMMA tiles per wave)

typedef float v2f __attribute__((ext_vector_type(2)));
typedef float v8f __attribute__((ext_vector_type(8)));

// ---------------------------------------------------------------------------
// Kernel 1: per-row norms, reciprocal norms (eps 1e-12 for contrastive), and
// cosine-similarity terms (eps 1e-8, clipped) for the similarity loss.
// ---------------------------------------------------------------------------
__global__ __launch_bounds__(256) void norms_kernel(
    const float* __restrict__ w, const float* __restrict__ o,
    float* __restrict__ rnw, float* __restrict__ rno,
    float* __restrict__ simterm) {
  const int row = blockIdx.x;
  const float* wr = w + (size_t)row * ND;
  const float* og = o + (size_t)row * ND;
  float w2 = 0.f, o2 = 0.f, wo = 0.f;
  for (int k = threadIdx.x; k < ND; k += 256) {
    float a = wr[k], b = og[k];
    w2 += a * a; o2 += b * b; wo += a * b;
  }
  for (int m = 16; m >= 1; m >>= 1) {
    w2 += __shfl_xor(w2, m);
    o2 += __shfl_xor(o2, m);
    wo += __shfl_xor(wo, m);
  }
  __shared__ float s0[8], s1[8], s2[8];
  const int wv = threadIdx.x >> 5, ln = threadIdx.x & 31;
  if (ln == 0) { s0[wv] = w2; s1[wv] = o2; s2[wv] = wo; }
  __syncthreads();
  if (threadIdx.x == 0) {
    float W = 0.f, O = 0.f, X = 0.f;
    for (int i = 0; i < 8; ++i) { W += s0[i]; O += s1[i]; X += s2[i]; }
    float nw = sqrtf(W), no = sqrtf(O);
    rnw[row] = 1.0f / fmaxf(nw, 1e-12f);
    rno[row] = 1.0f / fmaxf(no, 1e-12f);
    float c = X / (fmaxf(nw, 1e-8f) * fmaxf(no, 1e-8f));
    c = fminf(1.0f, fmaxf(-1.0f, c));
    simterm[row] = 1.0f - c;
  }
}

// ---------------------------------------------------------------------------
// Kernel 2: Gram + streaming fixed-shift logsumexp sums via fp32 WMMA.
// Block: 32 output rows x 8192 cols; 8 waves; each wave owns a 16-col tile
// and computes TWO stacked 16x16 tiles per B load (2 WMMAs per global b64).
// A (32 x 768) staged in LDS pre-scaled by rnorm[row]/T (~96.5 KB of the
// 320 KB WGP LDS). B streamed from L2 through an 8-deep register prefetch
// queue with compile-time slot indices (steady-state + tail split so no
// branches or register rotation appear in the hot loop). Deterministic.
// ---------------------------------------------------------------------------
__global__ __launch_bounds__(256) void gram_lse_kernel(
    const float* __restrict__ emb, const float* __restrict__ rnorm,
    const int* __restrict__ times, float* __restrict__ Sall,
    float* __restrict__ Spos, int* __restrict__ AnyP) {
  __shared__ float A_lds[ROWS * APITCH];
  __shared__ float redAll[8 * ROWS];
  __shared__ float redPos[8 * ROWS];
  __shared__ int   redAny[8 * ROWS];

  const int r0 = blockIdx.x * ROWS;
  const int tid = threadIdx.x;

  // Stage A tile, fold rnorm_row * (1/T) into it.
  for (int idx = tid; idx < ROWS * ND; idx += 256) {
    int m = idx / ND, k = idx - m * ND;
    A_lds[m * APITCH + k] =
        emb[(size_t)(r0 + m) * ND + k] * (rnorm[r0 + m] * 10.0f);
  }
  __syncthreads();

  const int wave = tid >> 5;
  const int lane = tid & 31;
  const int h = lane >> 4;   // half-wave: selects K pair / row group
  const int n = lane & 15;   // A row within tile == B column within tile

  int trow0[8], trow1[8];
#pragma unroll
  for (int k = 0; k < 8; ++k) {
    trow0[k] = times[r0 + k + 8 * h];
    trow1[k] = times[r0 + 16 + k + 8 * h];
  }

  float aAll0[8], aPos0[8], aAll1[8], aPos1[8];
  int anyM0 = 0, anyM1 = 0;
#pragma unroll
  for (int k = 0; k < 8; ++k) {
    aAll0[k] = 0.f; aPos0[k] = 0.f; aAll1[k] = 0.f; aPos1[k] = 0.f;
  }

  const float* Arow0 = &A_lds[n * APITCH + 2 * h];
  const float* Arow1 = Arow0 + 16 * APITCH;

  for (int jt = 0; jt < NB; jt += 128) {
    const int col = jt + wave * 16 + n;
    const float* bbase = emb + (size_t)col * ND + 2 * h;
    const float rn_c = rnorm[col];
    const int t_c = times[col];
    v8f d0 = {};
    v8f d1 = {};

    // 8-deep B prefetch queue (8 global b64 loads in flight).
    v2f bq[8];
#pragma unroll
    for (int i = 0; i < 8; ++i) bq[i] = *(const v2f*)(bbase + 4 * i);

    // Steady state: 184 iters (23 x unroll-8); slot index and prefetch are
    // compile-time within each unrolled body -> mov-free rotating registers.
#pragma unroll 8
    for (int k0 = 0; k0 < ND - 32; k0 += 4) {
      const int s = (k0 >> 2) & 7;
      v2f b = bq[s];
      bq[s] = *(const v2f*)(bbase + k0 + 32);
      v2f a0 = *(const v2f*)(Arow0 + k0);
      v2f a1 = *(const v2f*)(Arow1 + k0);
      d0 = __builtin_amdgcn_wmma_f32_16x16x4_f32(
          false, a0, false, b, (short)0, d0, false, false);
      d1 = __builtin_amdgcn_wmma_f32_16x16x4_f32(
          false, a1, false, b, (short)0, d1, false, false);
    }
    // Tail: drain the queue (no further loads).
#pragma unroll
    for (int k0 = ND - 32; k0 < ND; k0 += 4) {
      const int s = (k0 >> 2) & 7;
      v2f b = bq[s];
      v2f a0 = *(const v2f*)(Arow0 + k0);
      v2f a1 = *(const v2f*)(Arow1 + k0);
      d0 = __builtin_amdgcn_wmma_f32_16x16x4_f32(
          false, a0, false, b, (short)0, d0, false, false);
      d1 = __builtin_amdgcn_wmma_f32_16x16x4_f32(
          false, a1, false, b, (short)0, d1, false, false);
    }

#pragma unroll
    for (int k = 0; k < 8; ++k) {
      // tile 0: rows r0 .. r0+15
      {
        const int row = r0 + k + 8 * h;
        float sim = d0[k] * rn_c;           // = (z_i . z_j) / T
        float e = __expf(sim - 10.0f);
        const bool diag = (col == row);
        if (diag) e = 0.0f;
        aAll0[k] += e;
        int dt = trow0[k] - t_c; if (dt < 0) dt = -dt;
        if (!diag && dt < 365) { aPos0[k] += e; anyM0 |= (1 << k); }
      }
      // tile 1: rows r0+16 .. r0+31
      {
        const int row = r0 + 16 + k + 8 * h;
        float sim = d1[k] * rn_c;
        float e = __expf(sim - 10.0f);
        const bool diag = (col == row);
        if (diag) e = 0.0f;
        aAll1[k] += e;
        int dt = trow1[k] - t_c; if (dt < 0) dt = -dt;
        if (!diag && dt < 365) { aPos1[k] += e; anyM1 |= (1 << k); }
      }
    }
  }

  // Reduce across the 16 lanes of each half-wave (columns of the tile).
#pragma unroll
  for (int k = 0; k < 8; ++k) {
    for (int m = 8; m >= 1; m >>= 1) {
      aAll0[k] += __shfl_xor(aAll0[k], m);
      aPos0[k] += __shfl_xor(aPos0[k], m);
      aAll1[k] += __shfl_xor(aAll1[k], m);
      aPos1[k] += __shfl_xor(aPos1[k], m);
    }
  }
  for (int m = 8; m >= 1; m >>= 1) {
    anyM0 |= __shfl_xor(anyM0, m);
    anyM1 |= __shfl_xor(anyM1, m);
  }
  if (n == 0) {
#pragma unroll
    for (int k = 0; k < 8; ++k) {
      const int m0 = k + 8 * h;
      redAll[wave * ROWS + m0] = aAll0[k];
      redPos[wave * ROWS + m0] = aPos0[k];
      redAny[wave * ROWS + m0] = (anyM0 >> k) & 1;
      const int m1 = 16 + k + 8 * h;
      redAll[wave * ROWS + m1] = aAll1[k];
      redPos[wave * ROWS + m1] = aPos1[k];
      redAny[wave * ROWS + m1] = (anyM1 >> k) & 1;
    }
  }
  __syncthreads();
  if (tid < ROWS) {
    float sA = 0.f, sP = 0.f; int an = 0;
    for (int w2 = 0; w2 < 8; ++w2) {
      sA += redAll[w2 * ROWS + tid];
      sP += redPos[w2 * ROWS + tid];
      an |= redAny[w2 * ROWS + tid];
    }
    Sall[r0 + tid] = sA;
    Spos[r0 + tid] = sP;
    AnyP[r0 + tid] = an;
  }
}

// ---------------------------------------------------------------------------
// Kernel 3: Cox loss. Stable descending sort by time (key = t*8192+(8191-i))
// via single-workgroup bitonic sort in LDS, then two-level inclusive scan of
// exp(clip(lr)) for the at-risk cumulative sums. 1 block, 1024 threads.
// ---------------------------------------------------------------------------
__global__ __launch_bounds__(1024) void cox_kernel(
    const float* __restrict__ lr, const int* __restrict__ times,
    const int* __restrict__ cens, float* __restrict__ cox_out) {
  __shared__ unsigned int keys[NB];
  __shared__ float sums[1024];
  __shared__ float redv[32];
  __shared__ int   redc[32];

  const int tid = threadIdx.x;
  for (int i = tid; i < NB; i += 1024)
    keys[i] = ((unsigned)times[i] << 13) | (unsigned)(8191 - i);
  __syncthreads();

  // Bitonic sort, descending.
  for (int k = 2; k <= NB; k <<= 1) {
    for (int j = k >> 1; j > 0; j >>= 1) {
      for (int i = tid; i < NB; i += 1024) {
        const int l = i ^ j;
        if (l > i) {
          const unsigned a = keys[i], b = keys[l];
          const bool up = (i & k) == 0;
          if ((up && a < b) || (!up && a > b)) { keys[i] = b; keys[l] = a; }
        }
      }
      __syncthreads();
    }
  }

  // Each thread owns 8 consecutive sorted elements.
  float lrv[8], ev[8]; int cv[8];
  float tot = 0.f;
  const int base = tid * 8;
#pragma unroll
  for (int q = 0; q < 8; ++q) {
    const unsigned key = keys[base + q];
    const int idx = 8191 - (int)(key & 8191u);
    float v = lr[idx];
    v = fminf(10.0f, fmaxf(-10.0f, v));
    lrv[q] = v;
    ev[q] = expf(v);
    cv[q] = cens[idx];
    tot += ev[q];
  }
  sums[tid] = tot;
  __syncthreads();
  for (int off = 1; off < 1024; off <<= 1) {
    float v = 0.f;
    if (tid >= off) v = sums[tid - off];
    __syncthreads();
    sums[tid] += v;
    __syncthreads();
  }
  float run = sums[tid] - tot;  // exclusive prefix
  float acc = 0.f; int cnt = 0;
#pragma unroll
  for (int q = 0; q < 8; ++q) {
    run += ev[q];
    const float per = lrv[q] - logf(run + 1e-15f);
    if (cv[q] == 1) { acc += per; ++cnt; }
  }
  for (int m = 16; m >= 1; m >>= 1) {
    acc += __shfl_xor(acc, m);
    cnt += __shfl_xor(cnt, m);
  }
  const int wv = tid >> 5, ln = tid & 31;
  if (ln == 0) { redv[wv] = acc; redc[wv] = cnt; }
  __syncthreads();
  if (tid == 0) {
    float a = 0.f; int c = 0;
    for (int i = 0; i < 32; ++i) { a += redv[i]; c += redc[i]; }
    cox_out[0] = (c > 0) ? (-a / (float)c) : 0.0f;
  }
}

// ---------------------------------------------------------------------------
// Kernel 4: final deterministic reduction -> scalar loss.
// per_row = log(S_all) - log(S_pos) (fixed shift cancels); valid requires
// censor==1 && any positive. 1 block, 256 threads.
// ---------------------------------------------------------------------------
__global__ __launch_bounds__(256) void finalize_kernel(
    const float* __restrict__ simterm, const int* __restrict__ cens,
    const float* __restrict__ SallW, const float* __restrict__ SposW,
    const int* __restrict__ anyW,
    const float* __restrict__ SallO, const float* __restrict__ SposO,
    const int* __restrict__ anyO,
    const float* __restrict__ coxp, float* __restrict__ out) {
  const int tid = threadIdx.x;
  float aSim = 0.f, aW = 0.f, aO = 0.f;
  int cW = 0, cO = 0;
  for (int i = tid; i < NB; i += 256) {
    aSim += simterm[i];
    const int ce = cens[i];
    if (ce == 1 && anyW[i]) { aW += logf(SallW[i]) - logf(SposW[i]); ++cW; }
    if (ce == 1 && anyO[i]) { aO += logf(SallO[i]) - logf(SposO[i]); ++cO; }
  }
  for (int m = 16; m >= 1; m >>= 1) {
    aSim += __shfl_xor(aSim, m);
    aW += __shfl_xor(aW, m);
    aO += __shfl_xor(aO, m);
    cW += __shfl_xor(cW, m);
    cO += __shfl_xor(cO, m);
  }
  __shared__ float s[3][8];
  __shared__ int si[2][8];
  const int wv = tid >> 5, ln = tid & 31;
  if (ln == 0) {
    s[0][wv] = aSim; s[1][wv] = aW; s[2][wv] = aO;
    si[0][wv] = cW; si[1][wv] = cO;
  }
  __syncthreads();
  if (tid == 0) {
    float S = 0.f, W = 0.f, O = 0.f; int CW = 0, CO = 0;
    for (int i = 0; i < 8; ++i) {
      S += s[0][i]; W += s[1][i]; O += s[2][i];
      CW += si[0][i]; CO += si[1][i];
    }
    const float cw = (CW > 0) ? (W / (float)CW) : 0.0f;
    const float co = (CO > 0) ? (O / (float)CO) : 0.0f;
    const float simloss = S / (float)NB;
    out[0] = coxp[0] + 1.0f * simloss + 0.1f * 0.5f * (cw + co);
  }
}

// ---------------------------------------------------------------------------
extern "C" void kernel_launch(void* const* d_in, const int* in_sizes, int n_in,
                              void* d_out, int out_size, void* d_ws,
                              size_t ws_size, hipStream_t stream) {
  const float* log_risks = (const float*)d_in[0];
  const int*   times     = (const int*)d_in[1];
  const int*   censor    = (const int*)d_in[2];
  const float* wsi       = (const float*)d_in[3];
  const float* omic      = (const float*)d_in[4];
  float* out = (float*)d_out;

  char* p = (char*)d_ws;
  float* rnw     = (float*)p; p += NB * sizeof(float);
  float* rno     = (float*)p; p += NB * sizeof(float);
  float* simterm = (float*)p; p += NB * sizeof(float);
  float* SallW   = (float*)p; p += NB * sizeof(float);
  float* SposW   = (float*)p; p += NB * sizeof(float);
  int*   anyW    = (int*)p;   p += NB * sizeof(int);
  float* SallO   = (float*)p; p += NB * sizeof(float);
  float* SposO   = (float*)p; p += NB * sizeof(float);
  int*   anyO    = (int*)p;   p += NB * sizeof(int);
  float* coxp    = (float*)p; p += sizeof(float);

  norms_kernel<<<NB, 256, 0, stream>>>(wsi, omic, rnw, rno, simterm);
  gram_lse_kernel<<<NB / ROWS, 256, 0, stream>>>(wsi, rnw, times,
                                                 SallW, SposW, anyW);
  gram_lse_kernel<<<NB / ROWS, 256, 0, stream>>>(omic, rno, times,
                                                 SallO, SposO, anyO);
  cox_kernel<<<1, 1024, 0, stream>>>(log_risks, times, censor, coxp);
  finalize_kernel<<<1, 256, 0, stream>>>(simterm, censor, SallW, SposW, anyW,
                                         SallO, SposO, anyO, coxp, out);
}